// CrossNetwork_59004260713084
// MI455X (gfx1250) — compile-verified
//
#include <hip/hip_runtime.h>

typedef __attribute__((ext_vector_type(2))) float v2f;
typedef __attribute__((ext_vector_type(8))) float v8f;

#define N_FEAT 1024
#define N_LAYER 4
#define TILE_M 16

__global__ __launch_bounds__(32) void crossnet_wmma_kernel(
    const float* __restrict__ x,    // [B, 1024]
    const float* __restrict__ ww,   // [4, 1024]
    const float* __restrict__ wb,   // [4, 1024]
    float* __restrict__ out)        // [B, 1024]
{
    __shared__ float Bsum[N_FEAT];      // b0+b1+b2+b3
    __shared__ float Tld[TILE_M * 4];   // t_k per row
    __shared__ float Alpha[TILE_M];

    const int lane = threadIdx.x;                 // 0..31 (wave32)
    const long rowbase = (long)blockIdx.x * TILE_M;

    // ---- Stage 0: Bsum into LDS + row-independent scalars c1..c3 ----
    float acc1 = 0.f, acc2 = 0.f, acc3 = 0.f;
    for (int c = lane; c < N_FEAT; c += 32) {
        float b0 = wb[0 * N_FEAT + c];
        float b1 = wb[1 * N_FEAT + c];
        float b2 = wb[2 * N_FEAT + c];
        float b3 = wb[3 * N_FEAT + c];
        float B1 = b0;
        float B2 = B1 + b1;
        float B3 = B2 + b2;
        Bsum[c] = B3 + b3;
        acc1 += B1 * ww[1 * N_FEAT + c];
        acc2 += B2 * ww[2 * N_FEAT + c];
        acc3 += B3 * ww[3 * N_FEAT + c];
    }
    // wave32 butterfly reduction -> every lane holds c1,c2,c3
    #pragma unroll
    for (int off = 16; off >= 1; off >>= 1) {
        acc1 += __shfl_xor(acc1, off, 32);
        acc2 += __shfl_xor(acc2, off, 32);
        acc3 += __shfl_xor(acc3, off, 32);
    }

    // ---- Stage 1: T = X_tile(16x1024) @ W^T via V_WMMA_F32_16X16X4_F32 ----
    // A 16x4 f32 layout: lane L -> M = L&15, K = k + 2*(L>>4) + {0,1} (float2).
    // B 4x16 f32 mirrors it: lane L -> N = L&15, K = k + 2*(L>>4) + {0,1}.
    // Only D columns 0..3 are consumed; columns 4..15 may hold junk, so all
    // lanes load w_{n&3} unconditionally (columns are independent in D).
    const int m    = lane & 15;
    const int koff = (lane >> 4) * 2;
    const float* aptr = x  + (rowbase + m) * N_FEAT + koff;
    const float* bptr = ww + (long)(m & 3) * N_FEAT + koff;

    v8f acc = {};
    for (int k = 0; k < N_FEAT; k += 4) {
        v2f a = *(const v2f*)(aptr + k);
        v2f b = *(const v2f*)(bptr + k);
        acc = __builtin_amdgcn_wmma_f32_16x16x4_f32(
            /*neg_a=*/false, a, /*neg_b=*/false, b,
            /*c_mod=*/(short)0, acc, /*reuse_a=*/false, /*reuse_b=*/false);
    }

    // D layout: VGPR v, lane L -> row = v + 8*(L>>4), col = L&15.
    // t_k for rows 0..7 live in lanes 0..3; rows 8..15 in lanes 16..19.
    if ((lane & 15) < 4) {
        const int nn    = lane & 3;
        const int rbase = (lane >> 4) * 8;
        #pragma unroll
        for (int r = 0; r < 8; ++r) {
            Tld[(rbase + r) * 4 + nn] = acc[r];
        }
    }
    __syncthreads();

    // ---- Stage 2: per-row alpha recurrence (16 lanes) ----
    if (lane < TILE_M) {
        float t0 = Tld[lane * 4 + 0];
        float t1 = Tld[lane * 4 + 1];
        float t2 = Tld[lane * 4 + 2];
        float t3 = Tld[lane * 4 + 3];
        float alpha = 1.0f + t0;              // alpha1 = 1*(1+t0) + c0, c0 = 0
        alpha = alpha * (1.0f + t1) + acc1;   // alpha2
        alpha = alpha * (1.0f + t2) + acc2;   // alpha3
        alpha = alpha * (1.0f + t3) + acc3;   // alpha4
        Alpha[lane] = alpha;
    }
    __syncthreads();

    // ---- Stage 3: out = x0 * alpha + Bsum (coalesced float4, X hits L2) ----
    for (int r = 0; r < TILE_M; ++r) {
        const float alpha = Alpha[r];
        const float* xr   = x   + (rowbase + r) * N_FEAT;
        float*       orow = out + (rowbase + r) * N_FEAT;
        #pragma unroll 2
        for (int c0 = lane * 4; c0 < N_FEAT; c0 += 32 * 4) {
            float4 xv = *(const float4*)(xr + c0);
            float4 bs = *(const float4*)(&Bsum[c0]);
            float4 ov;
            ov.x = xv.x * alpha + bs.x;
            ov.y = xv.y * alpha + bs.y;
            ov.z = xv.z * alpha + bs.z;
            ov.w = xv.w * alpha + bs.w;
            *(float4*)(orow + c0) = ov;
        }
    }
}

extern "C" void kernel_launch(void* const* d_in, const int* in_sizes, int n_in,
                              void* d_out, int out_size, void* d_ws, size_t ws_size,
                              hipStream_t stream) {
    (void)n_in; (void)out_size; (void)d_ws; (void)ws_size;
    const float* x  = (const float*)d_in[0];   // [B, 1024] f32
    const float* ww = (const float*)d_in[1];   // [4, 1024] f32
    const float* wb = (const float*)d_in[2];   // [4, 1024] f32
    float* out = (float*)d_out;

    const int rows  = in_sizes[0] / N_FEAT;    // 16384
    const int tiles = rows / TILE_M;           // 1024

    crossnet_wmma_kernel<<<tiles, 32, 0, stream>>>(x, ww, wb, out);
}